// GraphSAGEConvLayer_48258252538106
// MI455X (gfx1250) — compile-verified
//
#include <hip/hip_runtime.h>

typedef float v2f __attribute__((ext_vector_type(2)));
typedef float v8f __attribute__((ext_vector_type(8)));

#define D 128  // D_IN == D_OUT == 128

// ---------------------------------------------------------------------------
// Kernel 1: zero the workspace (neigh_sum [N*128] + deg [N])
// ---------------------------------------------------------------------------
__global__ __launch_bounds__(256) void sage_zero(float* __restrict__ p, long n) {
    long i = (long)blockIdx.x * blockDim.x + threadIdx.x;
    long stride = (long)gridDim.x * blockDim.x;
    for (; i < n; i += stride) p[i] = 0.0f;
}

// ---------------------------------------------------------------------------
// Kernel 2: edge scatter — neigh_sum[dst] += h[src]; deg[dst] += 1
// One wave32 per edge; each lane handles 4 consecutive features (float4 load,
// 4x global_atomic_add_f32). All hot data is L2-resident (h = 51 MB < 192 MB).
// ---------------------------------------------------------------------------
__global__ __launch_bounds__(256) void sage_scatter(
    const float* __restrict__ h, const int* __restrict__ src,
    const int* __restrict__ dst, float* __restrict__ nsum,
    float* __restrict__ deg, int n_edges)
{
    int e = blockIdx.x * 8 + (threadIdx.x >> 5);
    if (e >= n_edges) return;
    int lane = threadIdx.x & 31;
    int s = src[e];
    int d = dst[e];
    float4 v = *((const float4*)(h + (size_t)s * D) + lane);
    float* np = nsum + (size_t)d * D + lane * 4;
    atomicAdd(np + 0, v.x);
    atomicAdd(np + 1, v.y);
    atomicAdd(np + 2, v.z);
    atomicAdd(np + 3, v.w);
    if (lane == 0) atomicAdd(deg + d, 1.0f);
}

// ---------------------------------------------------------------------------
// Kernel 3: fused dual GEMM + bias via V_WMMA_F32_16X16X4_F32 (wave32).
//   out = h @ W_self^T + (nsum * rdeg) @ W_neigh^T + b
// One wave per 16-row M tile; 8 n-tiles (16x16) accumulated in v8f regs.
// A fragment (16x4 f32):  lane<16 -> (m=m0+lane, k=k0,k0+1)
//                         lane>=16 -> (m=m0+lane-16, k=k0+2,k0+3)  => float2 load
// B fragment (4x16 f32):  B[k][n] = W[n][k] -> W row n, consecutive k => float2 load
// C/D (16x16 f32): vgpr r -> row m0 + r + 8*half, col n0 + (lane&15)
// ---------------------------------------------------------------------------
__global__ __launch_bounds__(128) void sage_gemm(
    const float* __restrict__ h, const float* __restrict__ nsum,
    const float* __restrict__ deg, const float* __restrict__ Wself,
    const float* __restrict__ Wneigh, const float* __restrict__ bias,
    float* __restrict__ out, int n_nodes)
{
    const int lane = threadIdx.x & 31;
    const int wave = threadIdx.x >> 5;
    const int tile = blockIdx.x * 4 + wave;        // 16-row tile index
    const int m0   = tile * 16;
    if (m0 >= n_nodes) return;                     // uniform per-wave exit

    const int half  = lane >> 4;                   // 0: lanes 0-15, 1: lanes 16-31
    const int l15   = lane & 15;
    const int mA    = m0 + l15;                    // A row owned by this lane
    const int kOffA = half * 2;                    // hi half covers k+2,k+3

    // per-lane reciprocal degree (constant across all K steps)
    float dg   = deg[mA];
    float rdeg = 1.0f / fmaxf(dg, 1.0f);

    v8f acc[8];
    #pragma unroll
    for (int t = 0; t < 8; ++t) {
        #pragma unroll
        for (int r = 0; r < 8; ++r) acc[t][r] = 0.0f;
    }

    const float* hrow = h    + (size_t)mA * D + kOffA;
    const float* srow = nsum + (size_t)mA * D + kOffA;

    #pragma unroll 1
    for (int pass = 0; pass < 2; ++pass) {
        const float* arow = pass ? srow : hrow;
        const float* W    = pass ? Wneigh : Wself;
        const float* wb0  = W + (size_t)l15 * D + kOffA;  // n-tile 0 base
        #pragma unroll 4
        for (int ks = 0; ks < D; ks += 4) {
            v2f a = *(const v2f*)(arow + ks);
            if (pass) { a.x *= rdeg; a.y *= rdeg; }
            #pragma unroll
            for (int t = 0; t < 8; ++t) {
                v2f bf = *(const v2f*)(wb0 + ks + (size_t)t * 16 * D);
                acc[t] = __builtin_amdgcn_wmma_f32_16x16x4_f32(
                    /*neg_a=*/false, a, /*neg_b=*/false, bf,
                    /*c_mod=*/(short)0, acc[t],
                    /*reuse_a=*/false, /*reuse_b=*/false);
            }
        }
    }

    // bias + store: element r of acc[t] -> out[m0 + 8*half + r][t*16 + l15]
    const int mBase = m0 + half * 8;
    #pragma unroll
    for (int t = 0; t < 8; ++t) {
        int n = t * 16 + l15;
        float bb = bias[n];
        float* op = out + (size_t)mBase * D + n;
        #pragma unroll
        for (int r = 0; r < 8; ++r) {
            op[(size_t)r * D] = acc[t][r] + bb;
        }
    }
}

// ---------------------------------------------------------------------------
// Launcher
// Inputs (setup_inputs order): h, src, dst, W_self, W_neigh, b
// ---------------------------------------------------------------------------
extern "C" void kernel_launch(void* const* d_in, const int* in_sizes, int n_in,
                              void* d_out, int out_size, void* d_ws, size_t ws_size,
                              hipStream_t stream) {
    const float* h      = (const float*)d_in[0];
    const int*   src    = (const int*)d_in[1];
    const int*   dst    = (const int*)d_in[2];
    const float* Wself  = (const float*)d_in[3];
    const float* Wneigh = (const float*)d_in[4];
    const float* bias   = (const float*)d_in[5];
    float*       out    = (float*)d_out;

    const int n_nodes = in_sizes[0] / D;
    const int n_edges = in_sizes[1];

    float* nsum = (float*)d_ws;
    float* deg  = nsum + (size_t)n_nodes * D;

    // 1) zero neigh_sum + deg
    long ztot = (long)n_nodes * D + n_nodes;
    sage_zero<<<4096, 256, 0, stream>>>((float*)d_ws, ztot);

    // 2) edge scatter (8 edges per 256-thread block)
    int sblocks = (n_edges + 7) / 8;
    sage_scatter<<<sblocks, 256, 0, stream>>>(h, src, dst, nsum, deg, n_edges);

    // 3) fused WMMA GEMM (+mean +bias); 4 waves/block, 16 rows per wave
    int n_tiles = (n_nodes + 15) / 16;
    int gblocks = (n_tiles + 3) / 4;
    sage_gemm<<<gblocks, 128, 0, stream>>>(h, nsum, deg, Wself, Wneigh, bias,
                                           out, n_nodes);
}